// FGN_29197187678343
// MI455X (gfx1250) — compile-verified
//
#include <hip/hip_runtime.h>

typedef __attribute__((ext_vector_type(2))) float v2f;
typedef __attribute__((ext_vector_type(8))) float v8f;
typedef __attribute__((ext_vector_type(4))) unsigned int u32x4_t;
typedef __attribute__((ext_vector_type(8))) int i32x8_t;
typedef __attribute__((ext_vector_type(4))) int i32x4_t;

#define NEG_BIG (-3.402823466e38f)

// ---------------------------------------------------------------------------
// Tensor Data Mover helper: load a 1D/2D tile (f32 elements) from global to
// LDS. D# packing per CDNA5 ISA §8.3/8.4. pad32: 1 dword of LDS padding every
// 32 dwords -> stride-33 LDS rows. 6-arg builtin on this toolchain.
// ---------------------------------------------------------------------------
__device__ __forceinline__ void tdm_load_to_lds(const void* gptr,
                                                unsigned lds_off,
                                                unsigned tile0,   // elems, contiguous dim
                                                unsigned tile1,   // rows (1 for 1D)
                                                unsigned long long stride0, // elems
                                                bool pad32) {
    unsigned long long ga = (unsigned long long)gptr;
    u32x4_t g0;
    g0.x = 1u;                                   // count=1, user mode
    g0.y = lds_off;                              // lds_addr
    g0.z = (unsigned)ga;                         // global_addr[31:0]
    g0.w = (unsigned)(ga >> 32) | (2u << 30);    // global_addr[56:32] | type=2

    unsigned flags = (2u << 16);                 // data_size = 4 bytes
    if (pad32) flags |= (1u << 20) | (4u << 22); // pad_enable, interval=32dw, amount=1dw

    unsigned long long td0 = tile0;              // tensor_dim0 = tile0 (no OOB)
    unsigned long long td1 = tile1;              // tensor_dim1 = tile1
    unsigned long long q0 = (unsigned long long)flags | ((td0 & 0xFFFFull) << 48);
    unsigned long long q1 = ((td0 >> 16) & 0xFFFFull) | ((td1 & 0xFFFFFFFFull) << 16) |
                            ((unsigned long long)(tile0 & 0xFFFFu) << 48);   // tile_dim0
    unsigned long long q2 = (unsigned long long)(tile1 & 0xFFFFu) |          // tile_dim1
                            ((stride0 & 0xFFFFFFFFull) << 32);               // stride lo
    unsigned long long q3 = (stride0 >> 32) & 0xFFFFull;                     // stride hi
    union { unsigned long long q[4]; i32x8_t v; } g1u;
    g1u.q[0] = q0; g1u.q[1] = q1; g1u.q[2] = q2; g1u.q[3] = q3;
    i32x4_t z4 = {0, 0, 0, 0};
    i32x8_t z8 = {0, 0, 0, 0, 0, 0, 0, 0};
    __builtin_amdgcn_tensor_load_to_lds(g0, g1u.v, z4, z4, z8, 0);
}

// ---------------------------------------------------------------------------
// Kernel 0: f[o,i,h,w] = sum_k gw[o,i,k] * basis[k,h,w]   (tiny)
// ---------------------------------------------------------------------------
__global__ void __launch_bounds__(256)
gen_filters_kernel(const float* __restrict__ basis,
                   const float* __restrict__ gw1,
                   const float* __restrict__ gw2,
                   float* __restrict__ f1,   // [32,3,3,3]  = 864
                   float* __restrict__ f2) { // [32,32,3,3] = 9216
    int tid = blockIdx.x * blockDim.x + threadIdx.x;
    int nthr = gridDim.x * blockDim.x;
    for (int e = tid; e < 32 * 3 * 9; e += nthr) {
        int o = e / 27, r = e % 27, i = r / 9, hw = r % 9;
        float s = 0.f;
#pragma unroll
        for (int k = 0; k < 3; ++k) s += gw1[(o * 3 + i) * 3 + k] * basis[k * 9 + hw];
        f1[e] = s;
    }
    for (int e = tid; e < 32 * 32 * 9; e += nthr) {
        int o = e / 288, r = e % 288, i = r / 9, hw = r % 9;
        float s = 0.f;
#pragma unroll
        for (int k = 0; k < 3; ++k) s += gw2[(o * 32 + i) * 3 + k] * basis[k * 9 + hw];
        f2[e] = s;
    }
}

// ---------------------------------------------------------------------------
// Kernel 1: conv1 (3->32ch) + 2x2 maxpool + relu. Block per image.
// Zero-padded LDS input (3x30x30), branch-free conv; OUTPUT is written in a
// zero-padded [32][16][16] layout so conv2's im2col needs no bounds masks and
// the whole per-image activation block is one contiguous TDM tile.
// ---------------------------------------------------------------------------
__global__ void __launch_bounds__(256)
conv1_pool_relu_kernel(const float* __restrict__ x,   // [B,3,28,28]
                       const float* __restrict__ f1,  // [32,3,3,3]
                       float* __restrict__ h1) {      // [B,32,16,16] padded
    __shared__ float xsp[3 * 30 * 30];  // 2700, zero-padded borders
    __shared__ float fs[32 * 27];       // 864
    const int img = blockIdx.x;
    const float* xin = x + (size_t)img * 2352;
    for (int i = threadIdx.x; i < 2700; i += 256) {
        int ic = i / 900, rem = i % 900, y = rem / 30, xx = rem % 30;
        float v = 0.f;
        if (y >= 1 && y <= 28 && xx >= 1 && xx <= 28)
            v = xin[ic * 784 + (y - 1) * 28 + (xx - 1)];
        xsp[i] = v;
    }
    for (int i = threadIdx.x; i < 864; i += 256) fs[i] = f1[i];
    __syncthreads();

    float* out = h1 + (size_t)img * 32 * 256;
    for (int e = threadIdx.x; e < 32 * 256; e += 256) {
        const int oc = e >> 8, yy = (e >> 4) & 15, xx = e & 15;
        float val = 0.f;
        if (yy >= 1 && yy <= 14 && xx >= 1 && xx <= 14) {
            const int ph = yy - 1, pw = xx - 1;
            const float* w = fs + oc * 27;
            float patch[3][4][4];
            const int pb = (ph * 2) * 30 + (pw * 2);
#pragma unroll
            for (int ic = 0; ic < 3; ++ic)
#pragma unroll
                for (int y = 0; y < 4; ++y)
#pragma unroll
                    for (int x4 = 0; x4 < 4; ++x4)
                        patch[ic][y][x4] = xsp[ic * 900 + pb + y * 30 + x4];
            float m = NEG_BIG;
#pragma unroll
            for (int dy = 0; dy < 2; ++dy)
#pragma unroll
                for (int dx = 0; dx < 2; ++dx) {
                    float acc = 0.f;
#pragma unroll
                    for (int ic = 0; ic < 3; ++ic)
#pragma unroll
                        for (int kh = 0; kh < 3; ++kh)
#pragma unroll
                            for (int kw = 0; kw < 3; ++kw)
                                acc += w[ic * 9 + kh * 3 + kw] * patch[ic][dy + kh][dx + kw];
                    m = fmaxf(m, acc);
                }
            val = fmaxf(m, 0.f);
        }
        out[e] = val;   // borders = 0
    }
}

// ---------------------------------------------------------------------------
// conv2 column-strip worker. K-axis reordered as k' = khw*32 + ic so every
// LDS address in the unrolled loop is (base register + compile-time immediate):
//   B: h1p[(pbase + 512*lhalf) + ic0*256 + r*256 + kh*16 + kw]
//   A: k-paired f2p layout -> one aligned ds_load_b64 per fragment
// Zero per-iteration VALU, ~3 live base registers.
// ---------------------------------------------------------------------------
__device__ __forceinline__ void conv2_strip(const float* h1p, const float* f2p,
                                            int nt, int lmod, int lhalf,
                                            v8f& c0, v8f& c1) {
    const int n = nt * 16 + lmod;           // pixel index == oh*14+ow
    const int oh = n / 14;
    const int ow = n % 14;
    const int pB = oh * 16 + ow + 512 * lhalf;   // B base (lhalf folded in)
    const int aB = lhalf * 64 + lmod * 2;        // A base in k-paired layout
#pragma unroll
    for (int khw = 0; khw < 9; ++khw) {
        const int kh = khw / 3, kw = khw % 3;    // constants
#pragma unroll
        for (int ic0 = 0; ic0 < 32; ic0 += 4) {
            const int kk = khw * 32 + ic0;       // k' group base (constant)
            // A fragments: pair (k'+2*lhalf, k'+2*lhalf+1) is contiguous
            v2f a0 = *(const v2f*)&f2p[aB + (kk >> 1) * 64];        // mt = 0
            v2f a1 = *(const v2f*)&f2p[aB + (kk >> 1) * 64 + 32];   // mt = 1
            v2f b;
            b.x = h1p[pB + ic0 * 256 + kh * 16 + kw];               // r = 0
            b.y = h1p[pB + ic0 * 256 + 256 + kh * 16 + kw];         // r = 1
            c0 = __builtin_amdgcn_wmma_f32_16x16x4_f32(
                     false, a0, false, b, (short)0, c0, false, false);
            c1 = __builtin_amdgcn_wmma_f32_16x16x4_f32(
                     false, a1, false, b, (short)0, c1, false, false);
        }
    }
}

__device__ __forceinline__ void conv2_store_strip(float* outs, int nt, int lmod,
                                                  int lhalf, const v8f& c0,
                                                  const v8f& c1) {
    const int n = nt * 16 + lmod;
#pragma unroll
    for (int v = 0; v < 8; ++v) {
        const int m = v + 8 * lhalf;           // C: M = vgpr + 8*laneHalf
        outs[m * 208 + n] = c0[v];
        outs[(m + 16) * 208 + n] = c1[v];
    }
}

// ---------------------------------------------------------------------------
// Kernel 2: conv2 as implicit GEMM with V_WMMA_F32_16X16X4_F32.
//   Per image: C[32 oc x 196 pix] = f2[32 x 288] * im2col[288 x 196]
//   Padded activations via one 1D TDM load; f2 stored k-paired+transposed in
//   LDS; 8 waves cover 13 pixel strips, each strip = 2 WMMAs per B gather.
//   Fused 2x2 maxpool + relu via LDS staging.
// ---------------------------------------------------------------------------
__global__ void __launch_bounds__(256)
conv2_pool_relu_wmma_kernel(const float* __restrict__ h1,  // [B,32,16,16] padded
                            const float* __restrict__ f2,  // [32,32,3,3]
                            float* __restrict__ h2) {      // [B,1568]
    __shared__ float h1p[32 * 256];   // 8192 floats, padded image
    __shared__ float f2p[288 * 32];   // k-paired transposed; reused as conv-out
    const int img = blockIdx.x;
    const int wave  = threadIdx.x >> 5;
    const int lane  = threadIdx.x & 31;
    const int lmod  = lane & 15;
    const int lhalf = lane >> 4;

    if (wave == 0)
        tdm_load_to_lds(h1 + (size_t)img * 8192, (unsigned)(uintptr_t)h1p,
                        8192, 1, 8192, false);
    // f2 flat [oc][ic][khw] -> k' = khw*32+ic, dest (k'/2)*64 + oc*2 + (k'&1)
    for (int i = threadIdx.x; i < 9216; i += 256) {
        int oc = i / 288, rem = i % 288, ic = rem / 9, khw = rem % 9;
        int kp = khw * 32 + ic;
        f2p[(kp >> 1) * 64 + oc * 2 + (kp & 1)] = f2[i];
    }
    if (wave == 0) __builtin_amdgcn_s_wait_tensorcnt(0);
    __syncthreads();

    v8f cA0 = {}, cA1 = {}, cB0 = {}, cB1 = {};
    conv2_strip(h1p, f2p, wave, lmod, lhalf, cA0, cA1);        // nt = 0..7
    if (wave < 5)                                              // nt = 8..12
        conv2_strip(h1p, f2p, wave + 8, lmod, lhalf, cB0, cB1);

    __syncthreads();                  // all waves done reading h1p / f2p
    float* outs = f2p;                // reuse as conv-out [32 oc][208]
    conv2_store_strip(outs, wave, lmod, lhalf, cA0, cA1);
    if (wave < 5)
        conv2_store_strip(outs, wave + 8, lmod, lhalf, cB0, cB1);
    __syncthreads();

    float* out = h2 + (size_t)img * 1568;
    for (int e = threadIdx.x; e < 1568; e += 256) {
        const int oc = e / 49, p = e % 49;
        const int ph = p / 7, pw = p % 7;
        const int n0 = (ph * 2) * 14 + pw * 2;
        const float a0 = outs[oc * 208 + n0];
        const float a1 = outs[oc * 208 + n0 + 1];
        const float a2 = outs[oc * 208 + n0 + 14];
        const float a3 = outs[oc * 208 + n0 + 15];
        out[e] = fmaxf(fmaxf(fmaxf(a0, a1), fmaxf(a2, a3)), 0.f);
    }
}

// ---------------------------------------------------------------------------
// Kernel 3: FC1 GEMM  C[4096,1024] = A[4096,1568] * W1[1568,1024] + b1
//   64x64 block tile, 8 waves (4x2), each wave a 16x32 strip: the A fragment
//   is loaded once and feeds two WMMAs. TDM double-buffered pipeline;
//   A tile uses TDM LDS-padding (1 dword / 32) for the stride-33 layout.
// ---------------------------------------------------------------------------
#define FC1_K 1568
#define FC1_N 1024
#define FC1_CHUNKS (FC1_K / 32)
__global__ void __launch_bounds__(256)
fc1_wmma_kernel(const float* __restrict__ A,
                const float* __restrict__ W1,
                const float* __restrict__ b1,
                float* __restrict__ C) {
    __shared__ float As[2][64 * 33];    // stride 33 (written padded by TDM)
    __shared__ float Bs[2][32 * 64];    // [k][n], row stride 64
    const int bm = blockIdx.x * 64;
    const int bn = blockIdx.y * 64;
    const int wave  = threadIdx.x >> 5;
    const int lane  = threadIdx.x & 31;
    const int wm = (wave >> 1) * 16;   // 0,16,32,48
    const int wn = (wave & 1) * 32;    // 0,32
    const int lmod = lane & 15, lhalf = lane >> 4;

    const unsigned as_off0 = (unsigned)(uintptr_t)&As[0][0];
    const unsigned as_off1 = (unsigned)(uintptr_t)&As[1][0];
    const unsigned bs_off0 = (unsigned)(uintptr_t)&Bs[0][0];
    const unsigned bs_off1 = (unsigned)(uintptr_t)&Bs[1][0];
    const int arow  = (wm + lmod) * 33 + 2 * lhalf;
    const int bcol0 = wn + lmod + 128 * lhalf;   // row stride 64 -> 2*lhalf rows
    const int bcol1 = bcol0 + 16;

    if (wave == 0) {
        tdm_load_to_lds(A + (size_t)bm * FC1_K, as_off0, 32, 64,
                        (unsigned long long)FC1_K, true);
        tdm_load_to_lds(W1 + (size_t)bn, bs_off0, 64, 32,
                        (unsigned long long)FC1_N, false);
    }

    v8f c0 = {}, c1 = {};
    for (int ch = 0; ch < FC1_CHUNKS; ++ch) {
        if (wave == 0) __builtin_amdgcn_s_wait_tensorcnt(0);
        __syncthreads();                       // chunk ch visible to all waves
        const int cur = ch & 1;
        if (wave == 0 && ch + 1 < FC1_CHUNKS) { // prefetch next chunk
            const int kc = (ch + 1) * 32;
            tdm_load_to_lds(A + (size_t)bm * FC1_K + kc, cur ? as_off0 : as_off1,
                            32, 64, (unsigned long long)FC1_K, true);
            tdm_load_to_lds(W1 + (size_t)kc * FC1_N + bn, cur ? bs_off0 : bs_off1,
                            64, 32, (unsigned long long)FC1_N, false);
        }
        const float* Asb = As[cur];
        const float* Bsb = Bs[cur];
#pragma unroll
        for (int k0 = 0; k0 < 32; k0 += 4) {
            v2f a, b0, b1v;
            a.x = Asb[arow + k0];
            a.y = Asb[arow + k0 + 1];
            b0.x = Bsb[bcol0 + k0 * 64];
            b0.y = Bsb[bcol0 + k0 * 64 + 64];
            b1v.x = Bsb[bcol1 + k0 * 64];
            b1v.y = Bsb[bcol1 + k0 * 64 + 64];
            c0 = __builtin_amdgcn_wmma_f32_16x16x4_f32(
                     false, a, false, b0, (short)0, c0, false, false);
            c1 = __builtin_amdgcn_wmma_f32_16x16x4_f32(
                     false, a, false, b1v, (short)0, c1, false, false);
        }
        __syncthreads();                       // buf reusable two chunks later
    }
    const int nn0 = bn + wn + lmod;
    const int nn1 = nn0 + 16;
    const float bias0 = b1[nn0];
    const float bias1 = b1[nn1];
#pragma unroll
    for (int v = 0; v < 8; ++v) {
        const int m = bm + wm + v + 8 * lhalf;
        C[(size_t)m * FC1_N + nn0] = c0[v] + bias0;
        C[(size_t)m * FC1_N + nn1] = c1[v] + bias1;
    }
}

// ---------------------------------------------------------------------------
// Kernel 4: FC2  out[4096,10] = A[4096,1024] * W2[1024,10] + b2
// ---------------------------------------------------------------------------
__global__ void __launch_bounds__(256)
fc2_kernel(const float* __restrict__ A,
           const float* __restrict__ W2,
           const float* __restrict__ b2,
           float* __restrict__ out) {
    const int wave = threadIdx.x >> 5;
    const int lane = threadIdx.x & 31;
    const int row = blockIdx.x * 8 + wave;
    const float* a = A + (size_t)row * 1024;
    float s[10];
#pragma unroll
    for (int c = 0; c < 10; ++c) s[c] = 0.f;
    for (int i = lane; i < 1024; i += 32) {
        const float av = a[i];
#pragma unroll
        for (int c = 0; c < 10; ++c) s[c] += av * W2[i * 10 + c];
    }
#pragma unroll
    for (int c = 0; c < 10; ++c) {
        float v = s[c];
#pragma unroll
        for (int off = 16; off > 0; off >>= 1) v += __shfl_down(v, off, 32);
        if (lane == 0) out[(size_t)row * 10 + c] = v + b2[c];
    }
}

// ---------------------------------------------------------------------------
extern "C" void kernel_launch(void* const* d_in, const int* in_sizes, int n_in,
                              void* d_out, int out_size, void* d_ws, size_t ws_size,
                              hipStream_t stream) {
    const float* x     = (const float*)d_in[0];  // [4096,3,28,28]
    const float* basis = (const float*)d_in[1];  // filters [3,3,3]
    const float* gw1   = (const float*)d_in[2];  // [32,3,3]
    const float* gw2   = (const float*)d_in[3];  // [32,32,3]
    const float* W1    = (const float*)d_in[4];  // [1568,1024]
    const float* b1    = (const float*)d_in[5];  // [1024]
    const float* W2    = (const float*)d_in[6];  // [1024,10]
    const float* b2    = (const float*)d_in[7];  // [10]
    float* out = (float*)d_out;                  // [4096,10]

    float* ws  = (float*)d_ws;
    float* f1  = ws;                              // 864
    float* f2  = ws + 864;                        // 9216
    float* h1  = ws + 10080;                      // 4096*8192 (padded [32][16][16])
    float* h2  = h1 + (size_t)4096 * 8192;        // 4096*1568
    float* fc1 = h2 + (size_t)4096 * 1568;        // 4096*1024

    gen_filters_kernel<<<8, 256, 0, stream>>>(basis, gw1, gw2, f1, f2);
    conv1_pool_relu_kernel<<<4096, 256, 0, stream>>>(x, f1, h1);
    conv2_pool_relu_wmma_kernel<<<4096, 256, 0, stream>>>(h1, f2, h2);
    dim3 g1(4096 / 64, 1024 / 64);
    fc1_wmma_kernel<<<g1, 256, 0, stream>>>(h2, W1, b1, fc1);
    fc2_kernel<<<512, 256, 0, stream>>>(fc1, W2, b2, out);
}